// ClassificationKNNLoss_74663711473733
// MI455X (gfx1250) — compile-verified
//
#include <hip/hip_runtime.h>
#include <hip/hip_bf16.h>
#include <math.h>

#define NPTS 8192
#define DIM  256
#define TK   16
#define PADD 258   // LDS row pitch for A block (floats), keeps float2 loads 8B-aligned

typedef __attribute__((ext_vector_type(2))) float v2f;
typedef __attribute__((ext_vector_type(8))) float v8f;

// Register-resident top-k (k=16, smallest). Fully unrolled so td/ti stay in VGPRs.
__device__ __forceinline__ void topk_insert(float (&td)[TK], int (&ti)[TK],
                                            float& tmax, float d, int c) {
    if (d < tmax) {
        // locate slot holding current max
        int pos = 0; float m = td[0];
#pragma unroll
        for (int e = 1; e < TK; ++e) { if (td[e] > m) { m = td[e]; pos = e; } }
        // replace it (constant-index selects after unroll -> stays in registers)
#pragma unroll
        for (int e = 0; e < TK; ++e) { if (e == pos) { td[e] = d; ti[e] = c; } }
        // recompute running max
        float nm = td[0];
#pragma unroll
        for (int e = 1; e < TK; ++e) nm = fmaxf(nm, td[e]);
        tmax = nm;
    }
}

// Kernel 1: squared row norms, one wave32 per row (8 floats/lane, shuffle reduce).
__global__ __launch_bounds__(256) void row_norms_kernel(const float* __restrict__ x,
                                                        float* __restrict__ norms) {
    int row  = (blockIdx.x * blockDim.x + threadIdx.x) >> 5;
    int lane = threadIdx.x & 31;
    if (row < NPTS) {
        const float4* p = (const float4*)(x + (size_t)row * DIM) + lane * 2;
        float4 a = p[0], b = p[1];
        float s = a.x*a.x + a.y*a.y + a.z*a.z + a.w*a.w
                + b.x*b.x + b.y*b.y + b.z*b.z + b.w*b.w;
#pragma unroll
        for (int m = 16; m >= 1; m >>= 1) s += __shfl_xor(s, m, 32);
        if (lane == 0) norms[row] = s;
    }
}

// Kernel 2: fused Gram(WMMA f32) -> distance -> denom + top-16 per row.
// One WG owns 16 rows; 8 waves stripe the 512 column tiles.
__global__ __launch_bounds__(256) void knn_main_kernel(const float* __restrict__ x,
                                                       const int* __restrict__ y,
                                                       const float* __restrict__ norms,
                                                       float* __restrict__ partials) {
    __shared__ float As[16 * PADD];        // 16-row A block of x
    __shared__ float Dt[8][16][17];        // per-wave 16x16 distance tile (padded)
    __shared__ float candD[16][256];       // merged top-k candidates: 16 lists x 16 entries
    __shared__ int   candI[16][256];
    __shared__ float dsumLds[16][16];      // per-(row,lane-slot) partial denom
    __shared__ float rn_s[16];
    __shared__ float rowMean[16];

    const int t    = threadIdx.x;
    const int lane = t & 31;
    const int w    = t >> 5;
    const int rowbase = blockIdx.x * 16;

    // Cooperative load of the 16x256 A block into LDS (padded pitch).
    {
        int r  = t >> 4;            // 0..15
        int c0 = (t & 15) * 16;     // 0..240
        const float4* src = (const float4*)(x + (size_t)(rowbase + r) * DIM + c0);
#pragma unroll
        for (int q = 0; q < 4; ++q) {
            float4 v = src[q];
            float* dst = &As[r * PADD + c0 + q * 4];
            dst[0] = v.x; dst[1] = v.y; dst[2] = v.z; dst[3] = v.w;
        }
        if (t < 16) rn_s[t] = norms[rowbase + t];
    }
    __syncthreads();

    const int r = lane & 15;     // WMMA: local col n / phase2: local row
    const int h = lane >> 4;     // half selector

    float td[TK]; int ti[TK];
#pragma unroll
    for (int e = 0; e < TK; ++e) { td[e] = INFINITY; ti[e] = -1; }
    float tmax = INFINITY;
    float dsum = 0.0f;
    const int myrow = rowbase + r;   // row this lane owns in phase 2

    for (int tile = w; tile < NPTS / 16; tile += 8) {
        const int colbase = tile * 16;

        // Prefetch next column tile's B rows for this wave (stride 8 tiles = 128 rows).
        if (tile + 8 < NPTS / 16) {
            __builtin_prefetch(x + (size_t)(colbase + 128 + r) * DIM + 2 * h, 0, 1);
        }

        // ---- 16x16 Gram tile via 64 chained V_WMMA_F32_16X16X4_F32 ----
        v8f acc = {0.f, 0.f, 0.f, 0.f, 0.f, 0.f, 0.f, 0.f};
        const float* abase = &As[r * PADD + 2 * h];                        // A frag: row r, K pair
        const float* bbase = x + (size_t)(colbase + r) * DIM + 2 * h;      // B frag: col r, K pair
#pragma unroll 8
        for (int kk = 0; kk < DIM / 4; ++kk) {
            v2f a = *(const v2f*)(abase + kk * 4);
            v2f b = *(const v2f*)(bbase + kk * 4);
            acc = __builtin_amdgcn_wmma_f32_16x16x4_f32(false, a, false, b,
                                                        (short)0, acc, false, false);
        }

        // ---- convert to Euclidean distance, park tile in LDS ----
        // Raw v_sqrt_f32 (~1 ulp) instead of libm's correctly-rounded expansion:
        // saves ~18 VALU ops per element on the critical path.
        float cnorm = norms[colbase + r];      // this lane's column norm (n = r)
#pragma unroll
        for (int v = 0; v < 8; ++v) {
            int m = v + 8 * h;
            float sq = rn_s[m] + cnorm - 2.0f * acc[v];
            Dt[w][m][r] = __builtin_amdgcn_sqrtf(fmaxf(sq, 0.0f));
        }
        __builtin_amdgcn_wave_barrier();

        // ---- phase 2: lane (r,h) scans row r, cols [8h, 8h+8) of this tile ----
#pragma unroll
        for (int j = 0; j < 8; ++j) {
            int cloc = h * 8 + j;
            int c    = colbase + cloc;
            float d  = Dt[w][r][cloc];
            if (c != myrow) {                 // exclude diagonal everywhere
                dsum += __expf(-d);
                topk_insert(td, ti, tmax, d, c);
            }
        }
        __builtin_amdgcn_wave_barrier();
    }

    // ---- deposit per-lane results: 16 candidate lists + denom partials per row ----
    int slot = w * 2 + h;                     // 0..15
    dsumLds[r][slot] = dsum;
#pragma unroll
    for (int e = 0; e < TK; ++e) {
        candD[r][slot * TK + e] = td[e];
        candI[r][slot * TK + e] = ti[e];
    }
    __syncthreads();

    // ---- final merge + loss, one lane per row ----
    if (t < 16) {
        int rr = t;
        float denom = 0.f;
#pragma unroll
        for (int s = 0; s < 16; ++s) denom += dsumLds[rr][s];

        float fd[TK]; int fi[TK];
#pragma unroll
        for (int e = 0; e < TK; ++e) { fd[e] = INFINITY; fi[e] = -1; }
        float fmx = INFINITY;
        for (int s2 = 0; s2 < 256; ++s2) {
            float d = candD[rr][s2];
            int   c = candI[rr][s2];
            if (c >= 0) topk_insert(fd, fi, fmx, d, c);
        }

        int   yr     = y[rowbase + rr];
        float logden = __logf(denom);
        float cnt = 0.f, sacc = 0.f;
#pragma unroll
        for (int e = 0; e < TK; ++e) {
            if (fi[e] >= 0 && y[fi[e]] == yr) {
                cnt  += 1.0f;
                sacc += (-fd[e] - logden);
            }
        }
        rowMean[rr] = (cnt > 0.f) ? (sacc / cnt) : 0.f;
    }
    __syncthreads();

    if (t == 0) {
        float sum = 0.f;
#pragma unroll
        for (int rr = 0; rr < 16; ++rr) sum += rowMean[rr];
        partials[blockIdx.x] = sum;
    }
}

// Kernel 3: deterministic final reduce of 512 per-WG partials.
__global__ void final_reduce_kernel(const float* __restrict__ partials,
                                    float* __restrict__ out) {
    float s = 0.f;
    for (int i = 0; i < NPTS / 16; ++i) s += partials[i];
    out[0] = -s / (float)NPTS;
}

extern "C" void kernel_launch(void* const* d_in, const int* in_sizes, int n_in,
                              void* d_out, int out_size, void* d_ws, size_t ws_size,
                              hipStream_t stream) {
    const float* x = (const float*)d_in[0];
    const int*   y = (const int*)d_in[1];
    float* out = (float*)d_out;
    float* ws  = (float*)d_ws;

    float* norms    = ws;           // 8192 floats
    float* partials = ws + NPTS;    // 512 floats

    row_norms_kernel<<<NPTS / 8, 256, 0, stream>>>(x, norms);
    knn_main_kernel<<<NPTS / 16, 256, 0, stream>>>(x, y, norms, partials);
    final_reduce_kernel<<<1, 1, 0, stream>>>(partials, out);
}